// ChannelAttentionLayer_20753281974539
// MI455X (gfx1250) — compile-verified
//
#include <hip/hip_runtime.h>

// out[b,d,n] = w[d] * x[b,d,n]
// B=8, D=2048, N=2048, fp32 in / fp32 out.
// Pure streaming op: ~268 MB of HBM traffic, ~11.5 us floor at 23.3 TB/s.
// Strategy: one 256-thread (8-wave, wave32) block per (b,d) row; 128-bit
// NT loads/stores (read-once / write-once data, working set > 192 MB L2);
// weight fetched once per block through the uniform/scalar path.

#define ROW_LEN   2048              // N (floats per row)
#define BLOCK_THR 256
#define VEC       4                 // float4
#define ROW_ITERS (ROW_LEN / (VEC * BLOCK_THR))   // = 2

// Native Clang vector type: accepted by __builtin_nontemporal_* and lowers
// to b128 vector memory ops.
typedef float v4f __attribute__((ext_vector_type(4)));

__global__ __launch_bounds__(BLOCK_THR) void channel_scale_kernel(
    const float* __restrict__ x,
    const float* __restrict__ w,
    float* __restrict__ y,
    int d_mask,        // D - 1 (D is a power of two)
    int nrows)         // B * D
{
    const int row = blockIdx.x;
    if (row >= nrows) return;

    // blockIdx.x is wave-uniform -> this lowers to a scalar (or
    // readfirstlane-broadcast) load; one weight fetch per block.
    const float s = w[row & d_mask];

    const size_t base = (size_t)row * ROW_LEN;
    const v4f* __restrict__ xv = reinterpret_cast<const v4f*>(x + base);
    v4f*       __restrict__ yv = reinterpret_cast<v4f*>(y + base);

#pragma unroll
    for (int i = 0; i < ROW_ITERS; ++i) {
        const int idx = (int)threadIdx.x + i * BLOCK_THR;
        // NT temporal hint: data is touched exactly once; don't pollute L2.
        v4f v = __builtin_nontemporal_load(xv + idx);
        v *= s;   // vector * scalar broadcast multiply
        __builtin_nontemporal_store(v, yv + idx);
    }
}

extern "C" void kernel_launch(void* const* d_in, const int* in_sizes, int n_in,
                              void* d_out, int out_size, void* d_ws, size_t ws_size,
                              hipStream_t stream) {
    (void)n_in; (void)d_ws; (void)ws_size;

    const float* x = (const float*)d_in[0];   // [B, D, N] fp32
    const float* w = (const float*)d_in[1];   // [D] fp32
    float* y       = (float*)d_out;           // [B, D, N] fp32

    const int D     = in_sizes[1];            // 2048 (power of two)
    const int nrows = out_size / ROW_LEN;     // B * D = 16384

    dim3 grid(nrows);
    dim3 block(BLOCK_THR);
    channel_scale_kernel<<<grid, block, 0, stream>>>(x, w, y, D - 1, nrows);
}